// ConvSBS_66881230733523
// MI455X (gfx1250) — compile-verified
//
#include <hip/hip_runtime.h>

typedef __attribute__((ext_vector_type(2))) float v2f;
typedef __attribute__((ext_vector_type(8))) float v8f;

#define NB      16
#define HH      64
#define WW      64
#define QIN     4
#define BOND    16
#define OUTQ    64
#define HPAD    65
#define WPAD    65
#define INTER   63                        // interior hp,wp in [1,63]
#define PIX_PER_IMG (INTER * INTER)       // 3969
#define NPIX    (NB * PIX_PER_IMG)        // 63504 = 3969 * 16
#define NBLK    (NPIX / 16)               // 3969 sixteen-pixel blocks
#define WAVES_PER_WG 8
#define NSLOT   97                        // 1 (W3) + 16 (W2) + 16 (W1) + 64 (W0)
#define S_W3    0
#define S_W2    1
#define S_W1    17
#define S_W0    33

static __device__ __forceinline__ v8f wmma4(v2f a, v2f b, v8f c) {
    // V_WMMA_F32_16X16X4_F32 : D(16x16 f32) = A(16x4 f32) * B(4x16 f32) + C
    return __builtin_amdgcn_wmma_f32_16x16x4_f32(
        /*neg_a=*/false, a, /*neg_b=*/false, b,
        /*c_mod=*/(short)0, c, /*reuse_a=*/false, /*reuse_b=*/false);
}

__global__ __launch_bounds__(256) void convsbs_zero(float4* __restrict__ o, int n4) {
    int i = blockIdx.x * blockDim.x + threadIdx.x;
    if (i < n4) o[i] = make_float4(0.f, 0.f, 0.f, 0.f);
}

__global__ __launch_bounds__(256) void convsbs_chain(
        const float* __restrict__ x,     // (B,64,64,4)
        const float* __restrict__ c0,    // (64,1,16,4)  -> W0[O][r0*4+q]
        const float* __restrict__ c1,    // (1,16,16,4)  -> W1[r0][s1*4+q]
        const float* __restrict__ c2,    // (1,16,16,4)  -> W2[s1][s2*4+q]
        const float* __restrict__ c3,    // (1,16,1,4)   -> W3[s2][q]
        float* __restrict__ out)         // (B,65,65,64)
{
    // Per-lane WMMA-A layout for a 16x4 f32 chunk:
    //   lane l (<16): M=l, K in {0,1} (VGPR0,VGPR1); lane l+16: M=l, K in {2,3}.
    // Stored per chunk as 32 lanes x 2 contiguous floats -> ds_load_b64.
    __shared__ float wlds[NSLOT * 64];

    const int tid = threadIdx.x;

    // ---- cooperative weight swizzle into LDS ----
    for (int i = tid; i < NSLOT * 64; i += 256) {
        int slot = i >> 6;
        int e    = i & 63;
        int lane = e >> 1;
        int v    = e & 1;
        int m    = lane & 15;
        int kk   = ((lane >> 4) << 1) + v;   // K index within chunk (0..3)
        float val;
        if (slot == S_W3) {
            val = c3[m * 4 + kk];                              // W3[m][kk]
        } else if (slot < S_W1) {
            int j = slot - S_W2;                               // K-chunk j of W2
            val = c2[m * 64 + 4 * j + kk];
        } else if (slot < S_W0) {
            int j = slot - S_W1;                               // K-chunk j of W1
            val = c1[m * 64 + 4 * j + kk];
        } else {
            int s = slot - S_W0;                               // W0 tile t, chunk j
            int t = s >> 4, j = s & 15;
            val = c0[(t * 16 + m) * 64 + 4 * j + kk];
        }
        wlds[i] = val;
    }
    __syncthreads();

    const int wave = tid >> 5;
    const int lane = tid & 31;
    const int blk  = blockIdx.x * WAVES_PER_WG + wave;   // wave-uniform
    if (blk >= NBLK) return;

    // ---- pixel decode (16 pixels per wave; this lane's pixel = lane&15) ----
    const int pix = blk * 16 + (lane & 15);
    const int b   = pix / PIX_PER_IMG;
    const int rr  = pix - b * PIX_PER_IMG;
    const int hp  = rr / INTER + 1;
    const int wp  = rr - (hp - 1) * INTER + 1;

    // ---- gather the 4 shifted q-vectors (float4 each) ----
    const float4* xv = (const float4*)x;                 // index (b*64+h)*64+w
    const int base = (b * HH + hp) * WW + wp;
    const float4 q3 = xv[base];                          // (hp  , wp  )
    const float4 q2 = xv[base - 1];                      // (hp  , wp-1)
    const float4 q1 = xv[base - WW];                     // (hp-1, wp  )
    const float4 q0 = xv[base - WW - 1];                 // (hp-1, wp-1)

    const int half = lane >> 4;
    // B-operand layout mirrors A: value at (VGPRv, lane) = coeff * x[q = 2*half + v]
    const float xb3a = half ? q3.z : q3.x, xb3b = half ? q3.w : q3.y;
    const float xb2a = half ? q2.z : q2.x, xb2b = half ? q2.w : q2.y;
    const float xb1a = half ? q1.z : q1.x, xb1b = half ? q1.w : q1.y;
    const float xb0a = half ? q0.z : q0.x, xb0b = half ? q0.w : q0.y;

    const v2f* wv = (const v2f*)wlds;

    // ---- stage 1: v3(16 x 16pix) = W3 @ X3 ----
    v2f a = wv[S_W3 * 32 + lane];
    v2f bb; bb[0] = xb3a; bb[1] = xb3b;
    v8f zeroAcc = {};
    v8f v3d = wmma4(a, bb, zeroAcc);

    // ---- stage 2: u = W2 @ (v3 ⊗ x2), K=64 in 16 chunks ----
    v8f u = {};
#pragma unroll
    for (int j = 0; j < 16; ++j) {
        // row j of v3 lives in VGPR (j&7), lane-half (j>>3); broadcast to this pixel's lane
        float row = __shfl(v3d[j & 7], (lane & 15) | ((j >> 3) << 4), 32);
        v2f bj; bj[0] = row * xb2a; bj[1] = row * xb2b;
        v2f aj = wv[(S_W2 + j) * 32 + lane];
        u = wmma4(aj, bj, u);
    }

    // ---- stage 3: v = W1 @ (u ⊗ x1) ----
    v8f vv = {};
#pragma unroll
    for (int j = 0; j < 16; ++j) {
        float row = __shfl(u[j & 7], (lane & 15) | ((j >> 3) << 4), 32);
        v2f bj; bj[0] = row * xb1a; bj[1] = row * xb1b;
        v2f aj = wv[(S_W1 + j) * 32 + lane];
        vv = wmma4(aj, bj, vv);
    }

    // ---- stage 4: out(64 x 16pix) = W0 @ (v ⊗ x0), 4 M-tiles ----
    v8f d0 = {}, d1 = {}, d2 = {}, d3 = {};
#pragma unroll
    for (int j = 0; j < 16; ++j) {
        float row = __shfl(vv[j & 7], (lane & 15) | ((j >> 3) << 4), 32);
        v2f bj; bj[0] = row * xb0a; bj[1] = row * xb0b;
        d0 = wmma4(wv[(S_W0 +  0 + j) * 32 + lane], bj, d0);
        d1 = wmma4(wv[(S_W0 + 16 + j) * 32 + lane], bj, d1);
        d2 = wmma4(wv[(S_W0 + 32 + j) * 32 + lane], bj, d2);
        d3 = wmma4(wv[(S_W0 + 48 + j) * 32 + lane], bj, d3);
    }

    // ---- store: D layout -> lane owns O = t*16 + 8*half + (0..7), pixel = lane&15 ----
    const int obase = ((b * HPAD + hp) * WPAD + wp) * OUTQ + 8 * half;
#define STORE_TILE(dt, t) {                                              \
        float* p = out + obase + (t) * 16;                               \
        ((float4*)p)[0] = make_float4(dt[0], dt[1], dt[2], dt[3]);       \
        ((float4*)p)[1] = make_float4(dt[4], dt[5], dt[6], dt[7]);       \
    }
    STORE_TILE(d0, 0)
    STORE_TILE(d1, 1)
    STORE_TILE(d2, 2)
    STORE_TILE(d3, 3)
#undef STORE_TILE
}

extern "C" void kernel_launch(void* const* d_in, const int* in_sizes, int n_in,
                              void* d_out, int out_size, void* d_ws, size_t ws_size,
                              hipStream_t stream) {
    (void)in_sizes; (void)n_in; (void)d_ws; (void)ws_size;
    const float* x  = (const float*)d_in[0];
    const float* c0 = (const float*)d_in[1];
    const float* c1 = (const float*)d_in[2];
    const float* c2 = (const float*)d_in[3];
    const float* c3 = (const float*)d_in[4];
    float* out = (float*)d_out;

    // zero the whole output (borders stay zero; interior overwritten below)
    int n4 = out_size / 4;
    convsbs_zero<<<(n4 + 255) / 256, 256, 0, stream>>>((float4*)out, n4);

    int grid = (NBLK + WAVES_PER_WG - 1) / WAVES_PER_WG;   // 497 workgroups
    convsbs_chain<<<grid, 256, 0, stream>>>(x, c0, c1, c2, c3, out);
}